// hyperbolicGCN_highfreq_77266461655827
// MI455X (gfx1250) — compile-verified
//
#include <hip/hip_runtime.h>
#include <math.h>

#define MIN_NORM 1e-5f
#define EPS_H    1e-7f

typedef float v2f __attribute__((ext_vector_type(2)));
typedef float v4f __attribute__((ext_vector_type(4)));
typedef float v8f __attribute__((ext_vector_type(8)));

__device__ __forceinline__ float softplus_f(float z) {
  return (z > 20.f) ? z : log1pf(expf(z));
}

// wave32 all-lanes sum
__device__ __forceinline__ float waveReduceSum(float v) {
#pragma unroll
  for (int m = 16; m; m >>= 1) v += __shfl_xor(v, m, 32);
  return v;
}

// ---------------- degree / norm ----------------
__global__ void k_deg_init(float* dis, int N) {
  int i = blockIdx.x * blockDim.x + threadIdx.x;
  if (i < N) dis[i] = 1.0f;  // self-loop contributes 1 to deg
}

__global__ void k_deg_count(const long long* __restrict__ ei, float* dis, long long E) {
  long long e = (long long)blockIdx.x * blockDim.x + threadIdx.x;
  if (e < E) {
    long long r = __builtin_nontemporal_load(&ei[e]);  // row index
    atomicAdd(&dis[r], 1.0f);
  }
}

__global__ void k_deg_rsqrt(float* dis, int N) {
  int i = blockIdx.x * blockDim.x + threadIdx.x;
  if (i < N) dis[i] = 1.0f / sqrtf(dis[i]);  // deg^(-0.5), deg >= 1
}

// ---------------- initial embedding: x -> proj(expmap0(proj_tan0([0,x]), c0), c0) ----------------
__global__ void k_init_embed(const float* __restrict__ xin, const float* __restrict__ curv,
                             float* __restrict__ xcur, int N) {
  int wid  = (blockIdx.x * blockDim.x + threadIdx.x) >> 5;  // one wave per node
  int lane = threadIdx.x & 31;
  if (wid >= N) return;
  float c0 = softplus_f(curv[0]);
  float K = 1.0f / c0, sK = sqrtf(K);
  const float* xr = xin + (size_t)wid * 127;
  float v[4]; float ss = 0.f;
#pragma unroll
  for (int k = 0; k < 4; ++k) {
    int j = lane * 4 + k;                 // output dim j (0..127)
    float t = (j == 0) ? 0.f : xr[j - 1]; // input feature j-1
    v[k] = t; ss += t * t;
  }
  ss = waveReduceSum(ss);
  float xn = fmaxf(sqrtf(ss), MIN_NORM);
  float f  = sK * sinhf(xn / sK) / xn;
  float first = sqrtf(fmaxf(K + f * f * ss, EPS_H));
  float4 o;
  o.x = (lane == 0) ? first : f * v[0];
  o.y = f * v[1]; o.z = f * v[2]; o.w = f * v[3];
  *(float4*)(xcur + (size_t)wid * 128 + lane * 4) = o;
}

// ---------------- xt = proj_tan0(logmap0(x, c_in)) ----------------
__global__ void k_logmap(const float* __restrict__ xcur, const float* __restrict__ curv, int l,
                         float* __restrict__ xt, int N) {
  int wid  = (blockIdx.x * blockDim.x + threadIdx.x) >> 5;
  int lane = threadIdx.x & 31;
  if (wid >= N) return;
  float cin = softplus_f(curv[l]);
  float sK = sqrtf(1.0f / cin);
  float4 xv = *(const float4*)(xcur + (size_t)wid * 128 + lane * 4);
  float x0 = __shfl(xv.x, 0, 32);
  float a = (lane == 0) ? 0.f : xv.x;
  float ss = a * a + xv.y * xv.y + xv.z * xv.z + xv.w * xv.w;
  ss = waveReduceSum(ss);
  float yn = fmaxf(sqrtf(ss), MIN_NORM);
  float th = fmaxf(x0 / sK, 1.0f + EPS_H);
  float g  = sK * acoshf(th) / yn;
  float4 o;
  o.x = (lane == 0) ? 0.f : g * xv.x;
  o.y = g * xv.y; o.z = g * xv.z; o.w = g * xv.w;
  *(float4*)(xt + (size_t)wid * 128 + lane * 4) = o;
}

// ---------------- gate dots via WMMA: [dotA|dotB] = xt(N,128) @ [wA|wB](128,2) ----------------
// One wave handles a 16-node tile. A tile = 16x4 f32 per ISA layout:
//   lanes 0-15: M=lane, VGPR0=K0, VGPR1=K1 ; lanes 16-31: M=lane-16, VGPR0=K2, VGPR1=K3
// B tile 4x16 (only N=0 (wA) and N=1 (wB) nonzero), same K striping, N=lane&15.
// B is built branch-free: all lanes load both w halves, then select via multipliers,
// so no EXEC-mask branching serializes the WMMA chain.
// D: VGPR r: lanes0-15 -> (M=r, N=lane), lanes16-31 -> (M=8+r, N=lane-16).
__global__ void k_gate_wmma(const float* __restrict__ xt, const float* __restrict__ gw, int l,
                            float* __restrict__ dotA, float* __restrict__ dotB, int N) {
  int wid  = (blockIdx.x * blockDim.x + threadIdx.x) >> 5;
  int lane = threadIdx.x & 31;
  int base = wid * 16;
  if (base >= N) return;
  const float* w = gw + (size_t)l * 256;  // w[0:128]=wA (applied at col), w[128:256]=wB (at row)
  if (base + 16 <= N) {
    int m  = lane & 15;
    int kh = (lane >> 4) * 2;
    const float* arow = xt + (size_t)(base + m) * 128 + kh;
    const float* wA = w + kh;
    const float* wB = w + 128 + kh;
    float selA = (m == 0) ? 1.f : 0.f;   // lane owns output column 0 (dotA)
    float selB = (m == 1) ? 1.f : 0.f;   // lane owns output column 1 (dotB)
    v8f acc = {};
#pragma unroll 4
    for (int kc = 0; kc < 128; kc += 4) {
      v2f A = *(const v2f*)(arow + kc);        // global_load_b64
      v2f a = *(const v2f*)(wA + kc);          // global_load_b64 (wA pair)
      v2f b = *(const v2f*)(wB + kc);          // global_load_b64 (wB pair)
      v2f B;
      B.x = selA * a.x + selB * b.x;
      B.y = selA * a.y + selB * b.y;
      acc = __builtin_amdgcn_wmma_f32_16x16x4_f32(false, A, false, B,
                                                  (short)0, acc, false, false);
    }
    int mo = (lane >> 4) * 8;
    if (m == 0) {
#pragma unroll
      for (int r = 0; r < 8; ++r) dotA[base + mo + r] = acc[r];
    } else if (m == 1) {
#pragma unroll
      for (int r = 0; r < 8; ++r) dotB[base + mo + r] = acc[r];
    }
  } else {
    // scalar tail (N not multiple of 16)
    int node = base + (lane & 15);
    if (node < N) {
      const float* xr = xt + (size_t)node * 128;
      const float* wv = w + ((lane >> 4) ? 128 : 0);
      float s = 0.f;
      for (int k = 0; k < 128; ++k) s += xr[k] * wv[k];
      if (lane < 16) dotA[node] = s; else dotB[node] = s;
    }
  }
}

// ---------------- zero aggr ----------------
__global__ void k_zero4(float4* p, long long n4) {
  long long i = (long long)blockIdx.x * blockDim.x + threadIdx.x;
  if (i < n4) p[i] = make_float4(0.f, 0.f, 0.f, 0.f);
}

// ---------------- edge scatter: aggr[col] += dis[row]*dis[col]*tanh(dotA[col]+dotB[row]+b) * xt[row] ----------------
__global__ void k_edge(const long long* __restrict__ ei, long long E, long long M,
                       const float* __restrict__ xt, const float* __restrict__ dis,
                       const float* __restrict__ dotA, const float* __restrict__ dotB,
                       const float* __restrict__ gb, int l,
                       float* __restrict__ aggr) {
  long long edge = (long long)blockIdx.x * 8 + (threadIdx.x >> 5);  // one wave per edge
  int lane = threadIdx.x & 31;
  if (edge >= M) return;
  long long r, c;
  if (edge < E) {
    r = __builtin_nontemporal_load(&ei[edge]);
    c = __builtin_nontemporal_load(&ei[E + edge]);
  } else {
    r = c = edge - E;  // self-loop
  }
  float s;
  if (lane == 0) {
    float alpha = tanhf(dotA[c] + dotB[r] + gb[l]);
    s = dis[r] * dis[c] * alpha;
  }
  s = __shfl(s, 0, 32);
  float4 xv = *(const float4*)(xt + (size_t)r * 128 + lane * 4);
  float* ag = aggr + (size_t)c * 128 + lane * 4;
  atomicAdd(ag + 0, s * xv.x);
  atomicAdd(ag + 1, s * xv.y);
  atomicAdd(ag + 2, s * xv.z);
  atomicAdd(ag + 3, s * xv.w);
}

// ---------------- node update: x = proj(expmap0(proj_tan0(relu(logmap0(proj(expmap0(aggr,ci),ci),ci))),co),co) ----------------
__global__ void k_node_update(const float* __restrict__ aggr, const float* __restrict__ curv, int l,
                              float* __restrict__ xcur, int N) {
  int wid  = (blockIdx.x * blockDim.x + threadIdx.x) >> 5;
  int lane = threadIdx.x & 31;
  if (wid >= N) return;
  float cin = softplus_f(curv[l]), cout = softplus_f(curv[l + 1]);
  float Ki = 1.0f / cin,  sKi = sqrtf(Ki);
  float Ko = 1.0f / cout, sKo = sqrtf(Ko);
  float4 av = *(const float4*)(aggr + (size_t)wid * 128 + lane * 4);
  float a0 = (lane == 0) ? 0.f : av.x;  // expmap0 ignores dim 0
  float ss1 = a0 * a0 + av.y * av.y + av.z * av.z + av.w * av.w;
  ss1 = waveReduceSum(ss1);
  // expmap0(.,c_in) + proj
  float xn1 = fmaxf(sqrtf(ss1), MIN_NORM);
  float f1  = sKi * sinhf(xn1 / sKi) / xn1;
  float first1 = sqrtf(fmaxf(Ki + f1 * f1 * ss1, EPS_H));
  // logmap0(.,c_in), then relu
  float yn1 = fmaxf(f1 * sqrtf(ss1), MIN_NORM);
  float g1  = sKi * acoshf(fmaxf(first1 / sKi, 1.0f + EPS_H)) / yn1;
  float h = g1 * f1;
  float v0 = fmaxf(h * a0,   0.f);
  float v1 = fmaxf(h * av.y, 0.f);
  float v2 = fmaxf(h * av.z, 0.f);
  float v3 = fmaxf(h * av.w, 0.f);
  float ss2 = v0 * v0 + v1 * v1 + v2 * v2 + v3 * v3;
  ss2 = waveReduceSum(ss2);
  // expmap0(.,c_out) + proj
  float xn2 = fmaxf(sqrtf(ss2), MIN_NORM);
  float f2  = sKo * sinhf(xn2 / sKo) / xn2;
  float first2 = sqrtf(fmaxf(Ko + f2 * f2 * ss2, EPS_H));
  float4 o;
  o.x = (lane == 0) ? first2 : f2 * v0;
  o.y = f2 * v1; o.z = f2 * v2; o.w = f2 * v3;
  *(float4*)(xcur + (size_t)wid * 128 + lane * 4) = o;
}

// ---------------- final: out = proj_tan0(logmap0(x, c_last)) ----------------
__global__ void k_final(const float* __restrict__ xcur, const float* __restrict__ curv, int L,
                        float* __restrict__ out, int N) {
  int wid  = (blockIdx.x * blockDim.x + threadIdx.x) >> 5;
  int lane = threadIdx.x & 31;
  if (wid >= N) return;
  float cl = softplus_f(curv[L]);
  float sK = sqrtf(1.0f / cl);
  float4 xv = *(const float4*)(xcur + (size_t)wid * 128 + lane * 4);
  float x0 = __shfl(xv.x, 0, 32);
  float a = (lane == 0) ? 0.f : xv.x;
  float ss = a * a + xv.y * xv.y + xv.z * xv.z + xv.w * xv.w;
  ss = waveReduceSum(ss);
  float yn = fmaxf(sqrtf(ss), MIN_NORM);
  float g  = sK * acoshf(fmaxf(x0 / sK, 1.0f + EPS_H)) / yn;
  v4f o;
  o.x = (lane == 0) ? 0.f : g * xv.x;
  o.y = g * xv.y; o.z = g * xv.z; o.w = g * xv.w;
  __builtin_nontemporal_store(o, (v4f*)(out + (size_t)wid * 128 + lane * 4));
}

extern "C" void kernel_launch(void* const* d_in, const int* in_sizes, int n_in,
                              void* d_out, int out_size, void* d_ws, size_t ws_size,
                              hipStream_t stream) {
  const float*     x    = (const float*)d_in[0];
  const long long* ei   = (const long long*)d_in[1];
  const float*     curv = (const float*)d_in[2];
  const float*     gw   = (const float*)d_in[3];
  const float*     gb   = (const float*)d_in[4];

  int L = in_sizes[2] - 1;                 // curvatures has L+1 entries
  int dim = in_sizes[3] / (2 * L);         // gate_w is (L, 1, 2*dim); dim == 128 here
  int D = dim - 1;
  int N = in_sizes[0] / D;
  long long E = in_sizes[1] / 2;
  long long M = E + (long long)N;

  // workspace layout (floats)
  size_t nf = (size_t)N * 128;
  float* ws   = (float*)d_ws;
  float* xcur = ws;
  float* xt   = xcur + nf;
  float* aggr = xt + nf;
  float* dis  = aggr + nf;
  float* dotA = dis + N;
  float* dotB = dotA + N;

  const int TPB = 256;
  int nodeBlocks = (N + 7) / 8;                       // one wave (32 lanes) per node
  int tiles = (N + 15) / 16;
  int gateBlocks = (tiles + 7) / 8;                   // one wave per 16-node WMMA tile
  long long edgeBlocks = (M + 7) / 8;                 // one wave per edge
  long long zero4 = (long long)N * 32;                // N*128/4 float4s

  // degrees -> dis = deg^(-1/2)
  k_deg_init<<<(N + TPB - 1) / TPB, TPB, 0, stream>>>(dis, N);
  k_deg_count<<<(int)((E + TPB - 1) / TPB), TPB, 0, stream>>>(ei, dis, E);
  k_deg_rsqrt<<<(N + TPB - 1) / TPB, TPB, 0, stream>>>(dis, N);

  // initial hyperboloid embedding
  k_init_embed<<<nodeBlocks, TPB, 0, stream>>>(x, curv, xcur, N);

  for (int l = 0; l < L; ++l) {
    k_logmap<<<nodeBlocks, TPB, 0, stream>>>(xcur, curv, l, xt, N);
    k_gate_wmma<<<gateBlocks, TPB, 0, stream>>>(xt, gw, l, dotA, dotB, N);
    k_zero4<<<(int)((zero4 + TPB - 1) / TPB), TPB, 0, stream>>>((float4*)aggr, zero4);
    k_edge<<<(int)edgeBlocks, TPB, 0, stream>>>(ei, E, M, xt, dis, dotA, dotB, gb, l, aggr);
    k_node_update<<<nodeBlocks, TPB, 0, stream>>>(aggr, curv, l, xcur, N);
  }

  k_final<<<nodeBlocks, TPB, 0, stream>>>(xcur, curv, L, (float*)d_out, N);
}